// CrystalGraphConvNet_55113020342974
// MI455X (gfx1250) — compile-verified
//
#include <hip/hip_runtime.h>
#include <hip/hip_bf16.h>

typedef float v2f __attribute__((ext_vector_type(2)));
typedef float v8f __attribute__((ext_vector_type(8)));

#define EPS 1e-5f
constexpr int N0 = 16, NA = 96, ORIG = 92, F = 64, K = 41, H = 128, NC = 3;
constexpr int TWO_F = 2 * F;        // 128
constexpr int CIN = 2 * F + K;      // 169
constexpr int KP = 44;              // K padded to multiple of 4
constexpr int M_SMALL = N0 * NA;          // 1536
constexpr int M_BIG = N0 * NA * NA;       // 147456

// ---- workspace layout (float offsets) ----
constexpr size_t OFF_FEA   = 0;                       // 1536*64
constexpr size_t OFF_A     = OFF_FEA + 98304;         // 1536*128
constexpr size_t OFF_B     = OFF_A   + 196608;        // 1536*128
constexpr size_t OFF_SUM   = OFF_B   + 196608;        // 1536*64
constexpr size_t OFF_STATS = OFF_SUM + 98304;         // 768 floats
//   stats: sum1[128]@0  sq1[128]@128  sum2[64]@256  sq2[64]@320
//          scale1[128]@384 shift1[128]@512 scale2[64]@640 shift2[64]@704
constexpr size_t OFF_G     = OFF_STATS + 768;         // 147456*128

__device__ __forceinline__ float softplusf_(float x) {
  // stable: max(x,0) + log1p(exp(-|x|)) == jax.nn.softplus
  return fmaxf(x, 0.f) + log1pf(expf(-fabsf(x)));
}
__device__ __forceinline__ float sigmoidf_(float x) { return 1.f / (1.f + expf(-x)); }

// ---------------- CDNA5 async global->LDS staging (ASYNCcnt path) -----------
// LDS operand = low 32 bits of the generic pointer (LDS aperture truncation,
// ISA 10.2).
__device__ __forceinline__ uint32_t lds_off_of(const void* p) {
  return (uint32_t)(uintptr_t)p;
}
__device__ __forceinline__ void async_g2l_b128(uint32_t lds_off, const void* g) {
  asm volatile("global_load_async_to_lds_b128 %0, %1, off"
               :: "v"(lds_off), "v"(g) : "memory");
}
__device__ __forceinline__ void async_g2l_b32(uint32_t lds_off, const void* g) {
  asm volatile("global_load_async_to_lds_b32 %0, %1, off"
               :: "v"(lds_off), "v"(g) : "memory");
}
__device__ __forceinline__ void wait_async_all() {
  asm volatile("s_wait_asynccnt 0x0" ::: "memory");
}

// Stage a ROWS x COLS fp32 tile (global row stride gstr, LDS row stride LSTR)
// with 16-byte async chunks. Requires COLS%4==0 and 16B-aligned rows.
template <int ROWS, int COLS, int LSTR, int NTHREADS>
__device__ __forceinline__ void async_stage_tile(float* lds, const float* g,
                                                 int gstr, int tid) {
  constexpr int CH = COLS / 4;
  for (int idx = tid; idx < ROWS * CH; idx += NTHREADS) {
    int r = idx / CH, q = (idx % CH) * 4;
    async_g2l_b128(lds_off_of(&lds[r * LSTR + q]), g + (size_t)r * gstr + q);
  }
}

// One wave computes a 16x16 fp32 tile with V_WMMA_F32_16X16X4_F32.
// A layout (16x4 per step): lanes 0-15 hold rows, vgpr pair = {K, K+1} (lanes>=16: {K+2,K+3})
// B layout (4x16): vgpr pair = rows {K, K+1} striped over lanes 0-15 (lanes>=16: {K+2,K+3})
template <int LDA, int LDB, int KTOT>
__device__ __forceinline__ v8f wmma_tile_f32(const float* As, const float* Bs,
                                             int l, int half, int col0) {
  v8f acc = {};
#pragma unroll
  for (int k0 = 0; k0 < KTOT; k0 += 4) {
    v2f a, b;
    a.x = As[l * LDA + k0 + 2 * half];
    a.y = As[l * LDA + k0 + 2 * half + 1];
    b.x = Bs[(k0 + 2 * half) * LDB + col0 + l];
    b.y = Bs[(k0 + 2 * half + 1) * LDB + col0 + l];
    acc = __builtin_amdgcn_wmma_f32_16x16x4_f32(false, a, false, b, (short)0, acc,
                                                false, false);
  }
  return acc;
}

// ---------------- embedding: fea = atom @ emb_W + emb_b  (1536x92)x(92x64) ----
__global__ void k_embed(const float* __restrict__ atom, const float* __restrict__ W,
                        const float* __restrict__ bias, float* __restrict__ fea) {
  __shared__ float Bs[ORIG * 68];   // B stride padded 64->68
  __shared__ float As[16 * ORIG];   // stride 92 (conflict-free)
  const int tid = threadIdx.x;      // 128 threads = 4 waves
  const int m0 = blockIdx.x * 16;
  async_stage_tile<ORIG, F, 68, 128>(Bs, W, F, tid);
  async_stage_tile<16, ORIG, ORIG, 128>(As, atom + (size_t)m0 * ORIG, ORIG, tid);
  wait_async_all();
  __syncthreads();
  const int wave = tid >> 5, lane = tid & 31;
  const int half = lane >> 4, l = lane & 15, col0 = wave * 16;
  v8f acc = wmma_tile_f32<ORIG, 68, ORIG>(As, Bs, l, half, col0);
  const int col = col0 + l;
#pragma unroll
  for (int v = 0; v < 8; ++v)
    fea[(size_t)(m0 + v + 8 * half) * F + col] = acc[v] + bias[col];
}

// ---------------- per-layer small GEMMs: fea @ W1 -> Abuf, fea @ W2 -> Bbuf --
__global__ void k_small_gemm(const float* __restrict__ fea, const float* __restrict__ convW,
                             int layer, float* __restrict__ Abuf, float* __restrict__ Bbuf) {
  __shared__ float Bs[F * 132];    // B stride padded 128->132
  __shared__ float As[16 * 68];    // A stride padded 64->68
  const int tid = threadIdx.x;     // 256 threads = 8 waves
  const int part = blockIdx.y;     // 0: W1 rows [0,64), 1: W2 rows [64,128)
  const int m0 = blockIdx.x * 16;
  const float* W = convW + (size_t)layer * CIN * TWO_F + (size_t)part * F * TWO_F;
  async_stage_tile<F, TWO_F, 132, 256>(Bs, W, TWO_F, tid);
  async_stage_tile<16, F, 68, 256>(As, fea + (size_t)m0 * F, F, tid);
  wait_async_all();
  __syncthreads();
  const int wave = tid >> 5, lane = tid & 31;
  const int half = lane >> 4, l = lane & 15, col0 = wave * 16;
  v8f acc = wmma_tile_f32<68, 132, F>(As, Bs, l, half, col0);
  float* out = part ? Bbuf : Abuf;
  const int col = col0 + l;
#pragma unroll
  for (int v = 0; v < 8; ++v)
    out[(size_t)(m0 + v + 8 * half) * TWO_F + col] = acc[v];
}

// ---------------- big GEMM: nbr @ W3, fused gated assembly + BN1 stats -------
__global__ void k_big_gemm(const float* __restrict__ nbr, const int* __restrict__ adj,
                           const float* __restrict__ convW, const float* __restrict__ convb,
                           int layer, const float* __restrict__ Abuf,
                           const float* __restrict__ Bbuf, float* __restrict__ gbuf,
                           float* __restrict__ stats) {
  __shared__ float Bs[KP * 132];   // 44 x 128 (padded), zero tail rows
  __shared__ float As[16 * KP];    // 16 x 44 (stride 44: conflict-free)
  const int tid = threadIdx.x;     // 256 threads = 8 waves
  const int m0 = blockIdx.x * 16;
  const float* W3 = convW + (size_t)layer * CIN * TWO_F + (size_t)TWO_F * TWO_F;
  // B rows are 512B-aligned -> b128 async chunks
  async_stage_tile<K, TWO_F, 132, 256>(Bs, W3, TWO_F, tid);
  // A (nbr) rows are 41 floats (4B-aligned only) -> b32 async copies
  for (int idx = tid; idx < 16 * K; idx += 256) {
    int r = idx / K, k = idx % K;
    async_g2l_b32(lds_off_of(&As[r * KP + k]), nbr + (size_t)(m0 + r) * K + k);
  }
  // zero the K padding (rows 41..43 of B, cols 41..43 of A) with plain LDS stores
  for (int idx = tid; idx < 3 * TWO_F; idx += 256)
    Bs[(K + idx / TWO_F) * 132 + (idx % TWO_F)] = 0.f;
  if (tid < 48) As[(tid / 3) * KP + K + (tid % 3)] = 0.f;
  wait_async_all();
  __syncthreads();
  const int wave = tid >> 5, lane = tid & 31;
  const int half = lane >> 4, l = lane & 15, col0 = wave * 16;
  v8f acc = wmma_tile_f32<KP, 132, KP>(As, Bs, l, half, col0);
  const float* cb = convb + (size_t)layer * TWO_F;
  const int col = col0 + l;
  float lsum = 0.f, lsq = 0.f;
#pragma unroll
  for (int v = 0; v < 8; ++v) {
    const int m = m0 + v + 8 * half;       // (b*96+i)*96 + j
    const int bi = m / NA;                 // b*96+i
    const float af = (float)adj[m];
    float g = acc[v] + cb[col] + Abuf[(size_t)bi * TWO_F + col] +
              af * Bbuf[(size_t)bi * TWO_F + col];
    gbuf[(size_t)m * TWO_F + col] = g;
    lsum += g;
    lsq += g * g;
  }
  lsum += __shfl_xor(lsum, 16, 32);
  lsq  += __shfl_xor(lsq, 16, 32);
  if (half == 0) {                          // lanes 0..15 own distinct channels
    atomicAdd(&stats[col], lsum);           // sum1
    atomicAdd(&stats[128 + col], lsq);      // sumsq1
  }
}

// ---------------- BN param kernels ------------------------------------------
__global__ void k_zero_stats(float* __restrict__ stats) {
  if (threadIdx.x < 384) stats[threadIdx.x] = 0.f;
}

__global__ void k_bn1_params(const float* __restrict__ bng, const float* __restrict__ bnb,
                             int layer, float* __restrict__ stats) {
  const int c = threadIdx.x;  // 128
  const float inv = 1.f / (float)M_BIG;
  float mean = stats[c] * inv;
  float var = stats[128 + c] * inv - mean * mean;
  float sc = bng[(size_t)layer * TWO_F + c] * rsqrtf(var + EPS);
  stats[384 + c] = sc;
  stats[512 + c] = bnb[(size_t)layer * TWO_F + c] - mean * sc;
}

// -------- apply BN1, sigmoid(filt)*softplus(core), sum over neighbors -------
__global__ void k_apply_sum(const float* __restrict__ gbuf, float* __restrict__ stats,
                            float* __restrict__ summed) {
  const int bi = blockIdx.x;      // 0..1535  (b*96+i)
  const int c = threadIdx.x;      // 0..63
  const float sc0 = stats[384 + c],      sh0 = stats[512 + c];
  const float sc1 = stats[384 + 64 + c], sh1 = stats[512 + 64 + c];
  const float* gb = gbuf + (size_t)bi * NA * TWO_F;
  float acc = 0.f;
  for (int j = 0; j < NA; ++j) {
    float xf = gb[(size_t)j * TWO_F + c] * sc0 + sh0;
    float xc = gb[(size_t)j * TWO_F + F + c] * sc1 + sh1;
    acc += sigmoidf_(xf) * softplusf_(xc);
  }
  summed[(size_t)bi * F + c] = acc;
  atomicAdd(&stats[256 + c], acc);
  atomicAdd(&stats[320 + c], acc * acc);
}

__global__ void k_bn2_params(const float* __restrict__ bng, const float* __restrict__ bnb,
                             int layer, float* __restrict__ stats) {
  const int c = threadIdx.x;  // 64
  const float inv = 1.f / (float)M_SMALL;
  float mean = stats[256 + c] * inv;
  float var = stats[320 + c] * inv - mean * mean;
  float sc = bng[(size_t)layer * F + c] * rsqrtf(var + EPS);
  stats[640 + c] = sc;
  stats[704 + c] = bnb[(size_t)layer * F + c] - mean * sc;
}

// ---------------- residual update: fea = softplus(fea + bn2(summed)) --------
__global__ void k_fea_update(float* __restrict__ fea, const float* __restrict__ summed,
                             const float* __restrict__ stats) {
  const int idx = blockIdx.x * 256 + threadIdx.x;
  if (idx >= M_SMALL * F) return;
  const int c = idx & (F - 1);
  float v = fea[idx] + summed[idx] * stats[640 + c] + stats[704 + c];
  fea[idx] = softplusf_(v);
}

// ---------------- head: mean-pool, softplus, fc, softplus, out --------------
__global__ void k_head(const float* __restrict__ fea, const float* __restrict__ fcW,
                       const float* __restrict__ fcb, const float* __restrict__ outW,
                       const float* __restrict__ outb, float* __restrict__ out) {
  __shared__ float sp[N0 * F];   // softplus(crys)
  __shared__ float hl[N0 * H];
  const int tid = threadIdx.x;   // 128
  for (int idx = tid; idx < N0 * F; idx += 128) {
    int b = idx / F, c = idx % F;
    float s = 0.f;
    for (int i = 0; i < NA; ++i) s += fea[((size_t)b * NA + i) * F + c];
    sp[idx] = softplusf_(s * (1.f / (float)NA));
  }
  __syncthreads();
  {
    const int hc = tid;
    for (int b = 0; b < N0; ++b) {
      float acc = fcb[hc];
      for (int c = 0; c < F; ++c) acc += sp[b * F + c] * fcW[(size_t)c * H + hc];
      hl[b * H + hc] = softplusf_(acc);
    }
  }
  __syncthreads();
  if (tid < N0) {
    float acc = outb[0];
    for (int hh = 0; hh < H; ++hh) acc += hl[tid * H + hh] * outW[hh];
    out[tid] = acc;
  }
}

extern "C" void kernel_launch(void* const* d_in, const int* in_sizes, int n_in,
                              void* d_out, int out_size, void* d_ws, size_t ws_size,
                              hipStream_t stream) {
  (void)in_sizes; (void)n_in; (void)out_size; (void)ws_size;
  const float* atom  = (const float*)d_in[0];
  const float* nbr   = (const float*)d_in[1];
  const int*   adj   = (const int*)d_in[2];
  const float* embW  = (const float*)d_in[3];
  const float* embb  = (const float*)d_in[4];
  const float* convW = (const float*)d_in[5];
  const float* convb = (const float*)d_in[6];
  const float* bn1g  = (const float*)d_in[7];
  const float* bn1b  = (const float*)d_in[8];
  const float* bn2g  = (const float*)d_in[9];
  const float* bn2b  = (const float*)d_in[10];
  const float* fcW   = (const float*)d_in[11];
  const float* fcb   = (const float*)d_in[12];
  const float* outW  = (const float*)d_in[13];
  const float* outb  = (const float*)d_in[14];
  float* out = (float*)d_out;

  float* w      = (float*)d_ws;
  float* fea    = w + OFF_FEA;
  float* Abuf   = w + OFF_A;
  float* Bbuf   = w + OFF_B;
  float* summed = w + OFF_SUM;
  float* stats  = w + OFF_STATS;
  float* gbuf   = w + OFF_G;

  k_embed<<<dim3(M_SMALL / 16), dim3(128), 0, stream>>>(atom, embW, embb, fea);
  for (int layer = 0; layer < NC; ++layer) {
    k_zero_stats<<<1, 384, 0, stream>>>(stats);
    k_small_gemm<<<dim3(M_SMALL / 16, 2), dim3(256), 0, stream>>>(fea, convW, layer,
                                                                  Abuf, Bbuf);
    k_big_gemm<<<dim3(M_BIG / 16), dim3(256), 0, stream>>>(nbr, adj, convW, convb, layer,
                                                           Abuf, Bbuf, gbuf, stats);
    k_bn1_params<<<1, 128, 0, stream>>>(bn1g, bn1b, layer, stats);
    k_apply_sum<<<dim3(M_SMALL), dim3(64), 0, stream>>>(gbuf, stats, summed);
    k_bn2_params<<<1, 64, 0, stream>>>(bn2g, bn2b, layer, stats);
    k_fea_update<<<dim3(M_SMALL * F / 256), dim3(256), 0, stream>>>(fea, summed, stats);
  }
  k_head<<<1, 128, 0, stream>>>(fea, fcW, fcb, outW, outb, out);
}